// GroupedQueryAttention_80032420594002
// MI455X (gfx1250) — compile-verified
//
#include <hip/hip_runtime.h>
#include <hip/hip_bf16.h>
#include <math.h>

typedef __attribute__((ext_vector_type(2))) float v2f;
typedef __attribute__((ext_vector_type(8))) float v8f;

constexpr int cH = 8, cG = 4, cGSZ = 2, cHD = 256;
constexpr int cWIN = 1024;
constexpr int cB = 16, cT = 4, cD = 2048, cNBLK = 257;
constexpr float cSCALE = 0.0625f;   // 256^-0.5
constexpr int KC = 256;             // K-chunk for GEMMs

static __device__ __forceinline__ v8f wmma_f32(v2f a, v2f b, v8f c) {
  // D = A(16x4) * B(4x16) + C, all f32 (exact vs f32 reference)
  return __builtin_amdgcn_wmma_f32_16x16x4_f32(false, a, false, b, (short)0, c,
                                               false, false);
}

// CDNA5 async global->LDS copy (bypasses VGPRs, tracked by ASYNCcnt).
// Generic LDS pointers keep the LDS byte offset in addr[31:0].
static __device__ __forceinline__ void async_g2l_b128(void* lds, const void* g) {
  unsigned loff = (unsigned)(uintptr_t)lds;
  unsigned long long ga = (unsigned long long)(uintptr_t)g;
  asm volatile("global_load_async_to_lds_b128 %0, %1, off"
               :: "v"(loff), "v"(ga) : "memory");
}
static __device__ __forceinline__ void wait_async0() {
  asm volatile("s_wait_asynccnt 0" ::: "memory");
}

static __device__ __forceinline__ float redmax16(float v) {
  v = fmaxf(v, __shfl_xor(v, 1, 32));
  v = fmaxf(v, __shfl_xor(v, 2, 32));
  v = fmaxf(v, __shfl_xor(v, 4, 32));
  v = fmaxf(v, __shfl_xor(v, 8, 32));
  return v;
}
static __device__ __forceinline__ float redsum16(float v) {
  v += __shfl_xor(v, 1, 32);
  v += __shfl_xor(v, 2, 32);
  v += __shfl_xor(v, 4, 32);
  v += __shfl_xor(v, 8, 32);
  return v;
}

// ---------------------------------------------------------------------------
// GEMM: C[64 x Ntot] = A[64 x 2048] * W^T,  W rows split across up to 3 ptrs
// grid = (4 Mtiles, Ntot/64), block = 128 (4 waves, one 16x16 tile each)
// dynamic LDS = (16*KC + 4*16*KC) floats = 80 KB
// ---------------------------------------------------------------------------
__global__ void gemm_rows64(const float* __restrict__ A,
                            const float* __restrict__ W0, int lim0,
                            const float* __restrict__ W1, int lim1,
                            const float* __restrict__ W2,
                            float* __restrict__ C, int Ntot) {
  extern __shared__ float sm[];
  float* sA = sm;              // 16 x KC
  float* sW = sm + 16 * KC;    // 4 x (16 x KC)

  const int tid  = threadIdx.x;
  const int lane = tid & 31;
  const int w    = tid >> 5;
  const int m0   = blockIdx.x * 16;
  const int n0   = blockIdx.y * 64 + w * 16;

  const float* Wp;
  int nr;
  if (n0 < lim0)      { Wp = W0; nr = n0; }
  else if (n0 < lim1) { Wp = W1; nr = n0 - lim0; }
  else                { Wp = W2; nr = n0 - lim1; }

  float* sWw = sW + w * 16 * KC;
  v8f c0 = {}, c1 = {};   // two accumulators break the WMMA RAW chain

  for (int kc = 0; kc < cD; kc += KC) {
    // A chunk: 16 rows x KC floats = 1024 float4, 128 threads -> 8 each
    for (int i = 0; i < 8; ++i) {
      int idx4 = tid + 128 * i;
      int row  = idx4 >> 6;
      int col  = (idx4 & 63) << 2;
      async_g2l_b128(sA + row * KC + col,
                     A + (size_t)(m0 + row) * cD + kc + col);
    }
    // W chunk for this wave's 16 output columns: 1024 float4 / 32 lanes
    for (int i = 0; i < 32; ++i) {
      int idx4 = lane + 32 * i;
      int row  = idx4 >> 6;
      int col  = (idx4 & 63) << 2;
      async_g2l_b128(sWw + row * KC + col,
                     Wp + (size_t)(nr + row) * cD + kc + col);
    }
    // prefetch next W chunk while this one is consumed
    if (kc + KC < cD)
      __builtin_prefetch(Wp + (size_t)(nr + (lane & 15)) * cD + kc + KC +
                             ((lane >> 4) << 6), 0, 1);
    wait_async0();
    __syncthreads();
    const int base = (lane & 15) * KC + ((lane >> 4) << 1);
    for (int kk = 0; kk < KC; kk += 8) {
      v2f a0 = *(const v2f*)(sA + base + kk);
      v2f b0 = *(const v2f*)(sWw + base + kk);
      v2f a1 = *(const v2f*)(sA + base + kk + 4);
      v2f b1 = *(const v2f*)(sWw + base + kk + 4);
      c0 = wmma_f32(a0, b0, c0);
      c1 = wmma_f32(a1, b1, c1);
    }
    __syncthreads();
  }

  v8f c = c0 + c1;
  const int col   = n0 + (lane & 15);
  const int rbase = m0 + ((lane >> 4) << 3);
#pragma unroll
  for (int r = 0; r < 8; ++r)
    C[(size_t)(rbase + r) * Ntot + col] = c[r];
}

// ---------------------------------------------------------------------------
// RMS-norm + RoPE for q (512 rows) and k (256 rows). One wave per row.
// ---------------------------------------------------------------------------
__global__ void rmsrope(const float* __restrict__ qkv,
                        const float* __restrict__ qscale,
                        const float* __restrict__ kscale,
                        const int* __restrict__ kv_lens,
                        float* __restrict__ qf,
                        float* __restrict__ kn) {
  const int wid  = blockIdx.x * 4 + (threadIdx.x >> 5);
  const int lane = threadIdx.x & 31;

  const float* src;
  const float* sc;
  float* dst;
  int bt;
  if (wid < cB * cT * cH) {                    // q row
    bt = wid / cH;
    int h = wid % cH;
    src = qkv + (size_t)bt * 4096 + h * cHD;
    sc  = qscale;
    dst = qf + (size_t)wid * cHD;
  } else {                                     // k row
    int idx = wid - cB * cT * cH;
    bt = idx / cG;
    int g = idx % cG;
    src = qkv + (size_t)bt * 4096 + 2048 + g * cHD;
    sc  = kscale;
    dst = kn + (size_t)idx * cHD;
  }
  const int b = bt / cT, t = bt % cT;
  const int pos = kv_lens[b] + t;

  float v[8];
  float ss = 0.f;
#pragma unroll
  for (int j = 0; j < 8; ++j) {
    v[j] = src[lane + 32 * j];
    ss += v[j] * v[j];
  }
#pragma unroll
  for (int m = 1; m < 32; m <<= 1) ss += __shfl_xor(ss, m, 32);
  const float inv = rsqrtf(ss * (1.0f / 256.0f) + 1e-6f);

  float y[8];
#pragma unroll
  for (int j = 0; j < 8; ++j)
    y[j] = v[j] * inv * (1.0f + sc[lane + 32 * j]);

#pragma unroll
  for (int j = 0; j < 4; ++j) {
    int d1 = lane + 32 * j;                               // 0..127
    float freq = __expf(-(float)d1 * (1.0f / 128.0f) * 9.210340371976184f);
    float ang  = (float)pos * freq;
    float sn, csv;
    __sincosf(ang, &sn, &csv);
    dst[d1]       = y[j] * csv - y[j + 4] * sn;
    dst[d1 + 128] = y[j + 4] * csv + y[j] * sn;
  }
}

// ---------------------------------------------------------------------------
// Attention: 1 WG per (b,g); 4 waves split key blocks, flash-style merge.
// dynamic LDS = 4096 (Q) + 4*4096 (K) + 4*4096 (V) + 4*256 (P) + 128 floats
// K/V tiles staged with async global->LDS; fresh tokens patched after wait.
// ---------------------------------------------------------------------------
__global__ void attn(const float* __restrict__ qf, const float* __restrict__ kn,
                     const float* __restrict__ qkv,
                     const float* __restrict__ kb, const float* __restrict__ vb,
                     const int* __restrict__ btab, const int* __restrict__ kvl,
                     float* __restrict__ out) {
  extern __shared__ float sm[];
  float* sQ    = sm;                 // 16 x 256 (rows 8..15 zero pad)
  float* sKall = sm + 4096;          // 4 x 16 x 256 (reused for acc merge)
  float* sVall = sKall + 4 * 4096;   // 4 x 16 x 256
  float* sPall = sVall + 4 * 4096;   // 4 x 16 x 16
  float* sML   = sPall + 4 * 256;    // m: [0..31], l: [32..63]

  const int b = blockIdx.x / cG, g = blockIdx.x % cG;
  const int tid = threadIdx.x, lane = tid & 31, w = tid >> 5;
  float* sK = sKall + w * 4096;
  float* sV = sVall + w * 4096;
  float* sP = sPall + w * 256;

  // --- load Q tile (16x256), rows = hh*4 + t for hh in {0,1}, pad 8..15 ---
  for (int i = 0; i < 8; ++i) {
    int idx4 = tid + 128 * i;
    int row  = idx4 >> 6;
    int doff = (idx4 & 63) << 2;
    float4 val = make_float4(0.f, 0.f, 0.f, 0.f);
    if (row < 8) {
      int hh = row >> 2, t = row & 3;
      val = *(const float4*)(qf +
            (size_t)((b * cT + t) * cH + g * cGSZ + hh) * cHD + doff);
    }
    *(float4*)(sQ + row * 256 + doff) = val;
  }
  __syncthreads();

  const int kvlen   = kvl[b];
  const int pos_max = kvlen + cT - 1;
  int s_lo = pos_max - cWIN + 1;
  if (s_lo < 0) s_lo = 0;
  const int tb_lo = s_lo >> 4, tb_hi = pos_max >> 4;

  float m[8], l[8];
  v8f acc[16];
#pragma unroll
  for (int r = 0; r < 8; ++r) { m[r] = -__builtin_inff(); l[r] = 0.f; }
#pragma unroll
  for (int n = 0; n < 16; ++n) acc[n] = (v8f){};

  for (int tb = tb_lo + w; tb <= tb_hi; tb += 4) {
    const int blk = btab[b * cNBLK + tb];
    const size_t base = ((size_t)blk * cG + g) * (16 * cHD);

    // --- async-stage K/V tile (contiguous 16 KB each) ---
    for (int i = 0; i < 32; ++i) {
      unsigned o = (unsigned)(i * 32 + lane) * 4;   // float index, 16B units
      async_g2l_b128(sK + o, kb + base + o);
      async_g2l_b128(sV + o, vb + base + o);
    }
    wait_async0();

    // --- patch fresh tokens over the cached rows (<=4 rows, uniform) ---
    {
      int lo = tb * 16;           if (lo < kvlen) lo = kvlen;
      int hi = tb * 16 + 15;      if (hi > pos_max) hi = pos_max;
      for (int kpos = lo; kpos <= hi; ++kpos) {
        int off = kpos - tb * 16;
        int tt  = kpos - kvlen;
        for (int j = 0; j < 2; ++j) {
          int doff = (j * 32 + lane) << 2;          // 64 float4 per row
          float4 k4 = *(const float4*)(kn +
                        (size_t)((b * cT + tt) * cG + g) * cHD + doff);
          float4 v4 = *(const float4*)(qkv +
                        (size_t)(b * cT + tt) * 4096 + 3072 + g * cHD + doff);
          *(float4*)(sK + off * 256 + doff) = k4;
          *(float4*)(sV + off * 256 + doff) = v4;
        }
      }
    }

    // --- S = Q * K^T : 64 f32 WMMAs over HD=256 (two accumulators) ---
    v8f s0 = (v8f){}, s1 = (v8f){};
    const int abase = (lane & 15) * 256 + ((lane >> 4) << 1);
    for (int kk = 0; kk < 256; kk += 8) {
      v2f a0 = *(const v2f*)(sQ + abase + kk);
      v2f b0 = *(const v2f*)(sK + abase + kk);
      v2f a1 = *(const v2f*)(sQ + abase + kk + 4);
      v2f b1 = *(const v2f*)(sK + abase + kk + 4);
      s0 = wmma_f32(a0, b0, s0);
      s1 = wmma_f32(a1, b1, s1);
    }
    v8f s = s0 + s1;

    // --- mask + online softmax (rows live in lanes 0..15) ---
#pragma unroll
    for (int r = 0; r < 8; ++r) {
      float scv = s[r] * cSCALE;
      int kpos = tb * 16 + (lane & 15);
      int t = r & 3;
      int pq = kvlen + t;
      bool valid = (lane < 16) && (kpos <= pq) && (kpos + cWIN > pq);
      scv = valid ? scv : -__builtin_inff();
      float rm = redmax16(scv);
      float p = 0.f;
      if (rm > -1e37f) {                 // uniform within each 16-lane half
        float mn   = fmaxf(m[r], rm);
        float corr = __expf(m[r] - mn);  // 0 when m[r] was -inf
        p = valid ? __expf(scv - mn) : 0.f;
        float rs = redsum16(p);
        l[r] = l[r] * corr + rs;
        m[r] = mn;
#pragma unroll
        for (int n = 0; n < 16; ++n) acc[n][r] *= corr;
      }
      sP[(r + ((lane >> 4) << 3)) * 16 + (lane & 15)] = p;
    }

    // --- acc += P * V : 64 f32 WMMAs (P transposed via LDS) ---
    const int pb = (lane & 15) * 16 + ((lane >> 4) << 1);
#pragma unroll
    for (int kc = 0; kc < 4; ++kc) {
      v2f a = *(const v2f*)(sP + pb + kc * 4);
      int vrow = kc * 4 + ((lane >> 4) << 1);
#pragma unroll
      for (int n = 0; n < 16; ++n) {
        v2f bb;
        bb.x = sV[vrow * 256 + n * 16 + (lane & 15)];
        bb.y = sV[(vrow + 1) * 256 + n * 16 + (lane & 15)];
        acc[n] = wmma_f32(a, bb, acc[n]);
      }
    }
  }

  // --- merge the 4 waves' partial (m, l, acc) ---
  if (lane == 0) {
#pragma unroll
    for (int r = 0; r < 8; ++r) {
      sML[w * 8 + r]      = m[r];
      sML[32 + w * 8 + r] = l[r];
    }
  }
#pragma unroll
  for (int n = 0; n < 16; ++n)
#pragma unroll
    for (int r = 0; r < 8; ++r)
      sK[(r + ((lane >> 4) << 3)) * 256 + n * 16 + (lane & 15)] = acc[n][r];
  __syncthreads();

  // rows 0..7 x 256 dims = 2048 outputs, 128 threads -> 16 each
  for (int i = 0; i < 16; ++i) {
    int idx = tid + 128 * i;
    int row = idx >> 8;
    int dd  = idx & 255;
    float mstar = -__builtin_inff();
    for (int ww = 0; ww < 4; ++ww) mstar = fmaxf(mstar, sML[ww * 8 + row]);
    float lstar = 0.f, astar = 0.f;
    for (int ww = 0; ww < 4; ++ww) {
      float e = __expf(sML[ww * 8 + row] - mstar);   // exp(-inf)=0 for idle waves
      lstar += sML[32 + ww * 8 + row] * e;
      astar += sKall[ww * 4096 + row * 256 + dd] * e;
    }
    int hh = row >> 2, t = row & 3;
    out[(size_t)((b * cT + t) * cH + g * cGSZ + hh) * cHD + dd] = astar / lstar;
  }
}

// ---------------------------------------------------------------------------
extern "C" void kernel_launch(void* const* d_in, const int* in_sizes, int n_in,
                              void* d_out, int out_size, void* d_ws,
                              size_t ws_size, hipStream_t stream) {
  const float* x   = (const float*)d_in[0];
  const float* Wq  = (const float*)d_in[1];
  const float* Wk  = (const float*)d_in[2];
  const float* Wv  = (const float*)d_in[3];
  const float* Wo  = (const float*)d_in[4];
  const float* qns = (const float*)d_in[5];
  const float* kns = (const float*)d_in[6];
  const float* kbl = (const float*)d_in[7];
  const float* vbl = (const float*)d_in[8];
  const int* btab  = (const int*)d_in[9];
  const int* kvl   = (const int*)d_in[10];
  float* outp      = (float*)d_out;

  float* ws       = (float*)d_ws;
  float* qkv_pre  = ws;                   // 64*4096 = 262144 floats
  float* q_final  = qkv_pre + 262144;     // 131072
  float* k_new    = q_final + 131072;     // 65536
  float* attn_o   = k_new + 65536;        // 131072  (total ~2.3 MB)

  const size_t gemm_lds = (size_t)(16 * KC + 4 * 16 * KC) * sizeof(float);
  const size_t attn_lds = (size_t)(4096 + 4 * 4096 + 4 * 4096 + 4 * 256 + 128)
                          * sizeof(float);

  // 1) fused QKV projection: [64 x 4096] = x * [Wq;Wk;Wv]^T
  gemm_rows64<<<dim3(4, 64), 128, gemm_lds, stream>>>(
      x, Wq, 2048, Wk, 3072, Wv, qkv_pre, 4096);

  // 2) RMS-norm + RoPE for q and k (v passes through qkv_pre)
  rmsrope<<<192, 128, 0, stream>>>(qkv_pre, qns, kns, kvl, q_final, k_new);

  // 3) windowed attention with paged KV (fresh tokens override the cache)
  attn<<<cB * cG, 128, attn_lds, stream>>>(
      q_final, k_new, qkv_pre, kbl, vbl, btab, kvl, attn_o);

  // 4) output projection: out[64 x 2048] = attn * Wo^T
  gemm_rows64<<<dim3(4, 32), 128, gemm_lds, stream>>>(
      attn_o, Wo, 2048, nullptr, 2048, nullptr, outp, 2048);
}